// LocalNeighborhood_66460323938749
// MI455X (gfx1250) — compile-verified
//
#include <hip/hip_runtime.h>

typedef __attribute__((ext_vector_type(2))) float v2f;
typedef __attribute__((ext_vector_type(8))) float v8f;

#define B_   8
#define S_   2048
#define D_   128
#define K_   16
#define QT   16                 // queries per workgroup (one WMMA M-tile)
#define CTN  (S_ / 16)          // 128 column tiles
#define RSTR 2052               // padded LDS row stride (floats): bank-conflict free

__global__ __launch_bounds__(256)
void knn_wmma_kernel(const float* __restrict__ pts,    // [B,S,3]
                     const int*   __restrict__ fidx,   // [B,S,1]
                     const float* __restrict__ attr,   // [B,S,D]
                     float* __restrict__ out_d,        // [B,S,K]
                     float* __restrict__ out_i,        // [B,S,K]
                     float* __restrict__ out_a)        // [B,S,K,D]
{
    extern __shared__ float smem[];
    float* d2s = smem;                    // 16 * RSTR floats (~128 KB)
    float* qs  = smem + 16 * RSTR;        // 16 floats: |q|^2 per query row
    float* nbD = qs + 16;                 // 256 floats: selected distances
    int*   nbI = (int*)(nbD + 256);       // 256 ints:   selected neighbor cols

    const int tid  = threadIdx.x;
    const int wave = tid >> 5;
    const int lane = tid & 31;
    const int lh   = lane & 15;           // M (or N) within tile
    const int kh   = lane >> 4;           // which K-half this lane holds

    const int b  = blockIdx.x >> 7;       // S/16 = 128 tiles per batch
    const int qt = blockIdx.x & 127;
    const int q0 = qt * QT;

    // ---- load this lane's query point, pack A matrix (16x4 f32) ----
    const size_t qrow = (size_t)b * S_ + q0 + lh;
    const float qx = pts[qrow * 3 + 0];
    const float qy = pts[qrow * 3 + 1];
    const float qz = pts[qrow * 3 + 2];
    const float sQ = qx * qx + qy * qy + qz * qz;
    v2f a;
    a.x = kh ? qz : qx;                   // VGPR0: K=0 (lo half) / K=2 (hi half)
    a.y = kh ? 0.f : qy;                  // VGPR1: K=1 (lo half) / K=3 (hi half)

    if (wave == 0 && lane < 16) qs[lane] = sQ;
    __syncthreads();

    // |q_m|^2 needed per D accumulator VGPR r: row = r + kh*8
    float smv[8];
    #pragma unroll
    for (int r = 0; r < 8; ++r) smv[r] = qs[r + kh * 8];

    // ---- phase 1: full 16x2048 d^2 block into LDS via WMMA f32 16x16x4 ----
    for (int ct = wave; ct < CTN; ct += 8) {
        const int n = ct * 16 + lh;
        const size_t prow = (size_t)b * S_ + n;
        const float px = pts[prow * 3 + 0];
        const float py = pts[prow * 3 + 1];
        const float pz = pts[prow * 3 + 2];
        const float sN = px * px + py * py + pz * pz;
        v2f bv;
        bv.x = kh ? pz : px;
        bv.y = kh ? 0.f : py;

        v8f c = {};
        c = __builtin_amdgcn_wmma_f32_16x16x4_f32(
                /*neg_a=*/false, a, /*neg_b=*/false, bv,
                /*c_mod=*/(short)0, c, /*reuse_a=*/false, /*reuse_b=*/false);

        #pragma unroll
        for (int r = 0; r < 8; ++r) {
            // d2[m][n] = |q|^2 + |p|^2 - 2*dot
            const float d2 = __builtin_fmaf(-2.f, c[r], smv[r] + sN);
            d2s[(r + kh * 8) * RSTR + n] = d2;
        }
    }
    __syncthreads();

    // ---- phase 2: stable top-16 selection, one thread per query row ----
    if (tid < QT) {
        const float* row = d2s + tid * RSTR;
        float bd[K_]; int bi[K_];
        #pragma unroll
        for (int k = 0; k < K_; ++k) { bd[k] = 3.4e38f; bi[k] = 0; }
        for (int col = 0; col < S_; ++col) {
            const float d = row[col];
            if (d < bd[K_ - 1]) {          // strict '<' => stable ties
                int j = K_ - 1;
                while (j > 0 && bd[j - 1] > d) {
                    bd[j] = bd[j - 1]; bi[j] = bi[j - 1]; --j;
                }
                bd[j] = d; bi[j] = col;
            }
        }
        #pragma unroll
        for (int k = 0; k < K_; ++k) {
            nbD[tid * K_ + k] = bd[k];
            nbI[tid * K_ + k] = bi[k];
        }
    }
    __syncthreads();

    // ---- phase 3a: distances + index distance (one (m,k) per thread) ----
    {
        const int m = tid >> 4, k = tid & 15;
        const int q = q0 + m;
        const int ni = nbI[tid];
        const size_t o = ((size_t)b * S_ + q) * K_ + k;
        out_d[o] = nbD[tid];
        int dd = fidx[(size_t)b * S_ + q] - fidx[(size_t)b * S_ + ni];
        if (dd < 0) dd = -dd;
        out_i[o] = (float)dd;
    }

    // ---- phase 3b: attribute gather, wave-coalesced float4 rows ----
    for (int chunk = 0; chunk < 32; ++chunk) {
        const int pair = chunk * 8 + wave;         // 0..255 = (m,k)
        const int m = pair >> 4, k = pair & 15;
        const int ni = nbI[pair];
        const float4* src =
            (const float4*)(attr + ((size_t)b * S_ + ni) * D_);
        float4* dst =
            (float4*)(out_a + (((size_t)b * S_ + q0 + m) * K_ + k) * D_);
        dst[lane] = src[lane];                     // 32 lanes x 16B = 512B row
    }
}

extern "C" void kernel_launch(void* const* d_in, const int* in_sizes, int n_in,
                              void* d_out, int out_size, void* d_ws, size_t ws_size,
                              hipStream_t stream) {
    const float* pts  = (const float*)d_in[0];   // first_point [B,S,3] f32
    const int*   fidx = (const int*)  d_in[1];   // first_index [B,S,1] int
    const float* attr = (const float*)d_in[2];   // attribute   [B,S,D] f32

    float* out_d = (float*)d_out;                          // [B,S,K,1]
    float* out_i = out_d + (size_t)B_ * S_ * K_;           // [B,S,K,1]
    float* out_a = out_d + (size_t)2 * B_ * S_ * K_;       // [B,S,K,D]

    const size_t shmem = (size_t)(16 * RSTR + 16 + 256) * sizeof(float)
                       + (size_t)256 * sizeof(int);        // ~133.4 KB < 320 KB

    dim3 grid(B_ * (S_ / QT));                             // 1024 workgroups
    knn_wmma_kernel<<<grid, 256, shmem, stream>>>(pts, fidx, attr,
                                                  out_d, out_i, out_a);
}